// SparseInputLayer_11158325035042
// MI455X (gfx1250) — compile-verified
//
#include <hip/hip_runtime.h>
#include <stdint.h>

#define N_DENSE    64
#define N_SAMPLES  121
#define N_CHANNELS 384
#define BATCH      2048

#define ROW_IN     (N_DENSE * (1 + N_SAMPLES))   // 7808 floats per input batch row
#define ROW_IN_V4  (ROW_IN / 4)                  // 1952 float4
#define TILE       (N_CHANNELS * N_SAMPLES)      // 46464 floats per output tile
#define TILE_V4    (TILE / 4)                    // 11616 float4
#define THREADS    512

// One workgroup per batch element.
//  LDS: stage[7808] f32 (async-loaded input row) + acc[46464] f32 (dense tile)
//       + s_off[64] precomputed channel*121 offsets.  ~217 KB of the 320 KB/WGP.
__global__ __launch_bounds__(THREADS, 1)
void SparseInputLayer_scatter_kernel(const float* __restrict__ in,
                                     float* __restrict__ out) {
    __shared__ __align__(16) float acc[TILE];
    __shared__ __align__(16) float stage[ROW_IN];
    __shared__ int s_off[N_DENSE];

    const int tid = threadIdx.x;
    const int b   = blockIdx.x;

    // ---- 1. async-stage the whole input row (indices + data) into LDS -------
    {
        const float4* src4 = (const float4*)(in + (size_t)b * ROW_IN);
        for (int j = tid; j < ROW_IN_V4; j += THREADS) {
            uint64_t ga = (uint64_t)(uintptr_t)(src4 + j);
            uint32_t la = (uint32_t)(uintptr_t)((float4*)stage + j);
            asm volatile("global_load_async_to_lds_b128 %0, %1, off"
                         :: "v"(la), "v"(ga) : "memory");
        }
    }

    // ---- 2. zero the dense accumulator tile while the DMA is in flight ------
    {
        float4  z    = make_float4(0.f, 0.f, 0.f, 0.f);
        float4* acc4 = (float4*)acc;
        for (int j = tid; j < TILE_V4; j += THREADS) acc4[j] = z;
    }

    // wait for this wave's async loads; barrier makes all staging visible
    asm volatile("s_wait_asynccnt 0" ::: "memory");
    __syncthreads();

    // ---- 3. decode indices -> LDS float-row offsets --------------------------
    if (tid < N_DENSE) {
        int c = (int)stage[tid];
        c = min(max(c, 0), N_CHANNELS - 1);
        s_off[tid] = c * N_SAMPLES;
    }
    __syncthreads();

    // ---- 4. scatter-add 64 rows x 121 samples via LDS float atomics ---------
    // (ds_add_f32 handles duplicate channel indices within a batch)
    for (int k = tid; k < N_DENSE * N_SAMPLES; k += THREADS) {
        int d = k / N_SAMPLES;
        int s = k - d * N_SAMPLES;
        atomicAdd(&acc[s_off[d] + s], stage[N_DENSE + k]);
    }
    __syncthreads();

    // ---- 5. stream the tile LDS -> HBM with async store-from-LDS (b128) -----
    {
        float4* dst4 = (float4*)(out + (size_t)b * TILE);
        for (int j = tid; j < TILE_V4; j += THREADS) {
            uint64_t ga = (uint64_t)(uintptr_t)(dst4 + j);
            uint32_t la = (uint32_t)(uintptr_t)((float4*)acc + j);
            asm volatile("global_store_async_from_lds_b128 %0, %1, off"
                         :: "v"(ga), "v"(la) : "memory");
        }
    }
    asm volatile("s_wait_asynccnt 0" ::: "memory");
}

extern "C" void kernel_launch(void* const* d_in, const int* in_sizes, int n_in,
                              void* d_out, int out_size, void* d_ws, size_t ws_size,
                              hipStream_t stream) {
    (void)in_sizes; (void)n_in; (void)d_ws; (void)ws_size; (void)out_size;
    const float* in  = (const float*)d_in[0];
    float*       out = (float*)d_out;
    SparseInputLayer_scatter_kernel<<<BATCH, THREADS, 0, stream>>>(in, out);
}